// MambaBlock_69449621176623
// MI455X (gfx1250) — compile-verified
//
#include <hip/hip_runtime.h>
#include <hip/hip_bf16.h>

// ---------------------------------------------------------------------------
// Mamba block for MI455X (gfx1250, wave32).
//  - All GEMMs on V_WMMA_F32_16X16X4_F32 (fp32 tensor path == reference
//    precision; GEMMs total ~8 GFLOP, memory-bound at 23.3 TB/s).
//  - Selective scan (the latency-bound critical path): LDS double-buffered
//    chunks streamed with GLOBAL_LOAD_ASYNC_TO_LDS (ASYNCcnt) so the 4096
//    sequential steps read from LDS, not HBM. shfl_xor contraction over the
//    16 state lanes; D*x skip + SiLU gate fused into the store.
// ---------------------------------------------------------------------------

typedef __attribute__((ext_vector_type(2))) float v2f;
typedef __attribute__((ext_vector_type(8))) float v8f;

#define B_SZ     4
#define LSEQ     4096          // 64*64
#define CDIM     192
#define DI       384
#define NSTATE   16
#define DTRANK   12
#define MROWS    (B_SZ * LSEQ) // 16384
#define LDDBC    48            // 44 padded to 48

// scan chunking
#define T_CH     64                      // timesteps per LDS chunk
#define CHUNK_F  (T_CH * 32 + 3 * T_CH * 16)  // 5120 floats = 20KB / buffer

static __device__ inline v8f wmma_f32(v2f a, v2f b, v8f c) {
  // 8-arg form: (neg_a, A, neg_b, B, c_mod, C, reuse_a, reuse_b)
  return __builtin_amdgcn_wmma_f32_16x16x4_f32(
      false, a, false, b, (short)0, c, false, false);
}

static __device__ inline v8f zero8() {
  v8f r;
#pragma unroll
  for (int i = 0; i < 8; ++i) r[i] = 0.f;
  return r;
}

// -------- CDNA5 async global->LDS copy (ASYNCcnt), with safe fallback ------
static __device__ __forceinline__ void async_copy_f32(const float* g, float* l) {
#if __has_builtin(__builtin_amdgcn_global_load_async_to_lds_b32)
  __builtin_amdgcn_global_load_async_to_lds_b32(
      (__attribute__((address_space(1))) int*)g,
      (__attribute__((address_space(3))) int*)l, 0, 0);
#else
  *l = *g;                 // fallback: VGPR round trip (DS stores)
#endif
}

static __device__ __forceinline__ void wait_async_all() {
#if __has_builtin(__builtin_amdgcn_s_wait_asynccnt)
  __builtin_amdgcn_s_wait_asynccnt(0);
#endif
}

// ---------------------------------------------------------------------------
// K1: [B,C,H,W] -> [B*L, C] transpose + LayerNorm.  One wave per row.
// ---------------------------------------------------------------------------
__global__ void ln_transpose_kernel(const float* __restrict__ x,
                                    const float* __restrict__ lnw,
                                    const float* __restrict__ lnb,
                                    float* __restrict__ xn) {
  const int row  = blockIdx.x * (blockDim.x >> 5) + (threadIdx.x >> 5);
  const int lane = threadIdx.x & 31;
  if (row >= MROWS) return;
  const int b = row >> 12;          // /4096
  const int l = row & 4095;
  const float* xb = x + (size_t)b * CDIM * LSEQ + l;

  float v[6], s = 0.f, s2 = 0.f;
#pragma unroll
  for (int i = 0; i < 6; ++i) {
    const int c = lane + 32 * i;
    const float t = xb[(size_t)c * LSEQ];
    v[i] = t; s += t; s2 += t * t;
  }
#pragma unroll
  for (int off = 16; off > 0; off >>= 1) {
    s  += __shfl_xor(s,  off, 32);
    s2 += __shfl_xor(s2, off, 32);
  }
  const float mu   = s * (1.f / CDIM);
  const float var  = s2 * (1.f / CDIM) - mu * mu;
  const float rstd = rsqrtf(var + 1e-5f);
#pragma unroll
  for (int i = 0; i < 6; ++i) {
    const int c = lane + 32 * i;
    xn[(size_t)row * CDIM + c] = (v[i] - mu) * rstd * lnw[c] + lnb[c];
  }
}

// ---------------------------------------------------------------------------
// Generic WMMA GEMM:  C[M,N] = act( A[M,K] @ W[N,K]^T + bias )
// Wave tile: 16 x (16*NT).  ACT: 0 = none, 1 = softplus(+bias).
// ---------------------------------------------------------------------------
template <int NT, int ACT>
__global__ void gemm_wmma_kernel(const float* __restrict__ A,
                                 const float* __restrict__ W,
                                 float* __restrict__ C,
                                 const float* __restrict__ bias,
                                 int M, int N, int K, int lda, int ldc) {
  const int wave = blockIdx.x * (blockDim.x >> 5) + (threadIdx.x >> 5);
  const int lane = threadIdx.x & 31;
  const int ntiles = (N + 16 * NT - 1) / (16 * NT);
  const int m0 = (wave / ntiles) * 16;
  const int n0 = (wave % ntiles) * (16 * NT);
  if (m0 >= M) return;

  const int hf  = lane >> 4;   // half-wave selects K pair (ISA 16x4 A layout)
  const int l16 = lane & 15;
  const int rowA = m0 + l16;

  v8f acc[NT];
#pragma unroll
  for (int j = 0; j < NT; ++j) acc[j] = zero8();

  for (int k0 = 0; k0 < K; k0 += 4) {
    const int ka = k0 + 2 * hf;
    const v2f afrag = *(const v2f*)(A + (size_t)rowA * lda + ka);
#pragma unroll
    for (int j = 0; j < NT; ++j) {
      const int col = n0 + 16 * j + l16;
      v2f bfrag = {0.f, 0.f};
      if (col < N) bfrag = *(const v2f*)(W + (size_t)col * K + ka);
      acc[j] = wmma_f32(afrag, bfrag, acc[j]);
    }
  }

#pragma unroll
  for (int j = 0; j < NT; ++j) {
    const int col = n0 + 16 * j + l16;
    if (col >= N) continue;
    float bv = (ACT == 1) ? bias[col] : 0.f;
#pragma unroll
    for (int r = 0; r < 8; ++r) {
      const int row = m0 + r + 8 * hf;
      float v = acc[j][r];
      if (ACT == 1) {               // softplus(v + b)
        v += bv;
        v = (v > 20.f) ? v : __logf(1.f + __expf(v));
      }
      C[(size_t)row * ldc + col] = v;
    }
  }
}

// ---------------------------------------------------------------------------
// K3: causal depthwise conv1d (d_conv=4) over sequence + SiLU.
// One thread per (m, d).  xm = xz[:, 0:DI].
// ---------------------------------------------------------------------------
__global__ void conv_silu_kernel(const float* __restrict__ xz,
                                 const float* __restrict__ cw,
                                 const float* __restrict__ cb,
                                 float* __restrict__ xc) {
  const int idx = blockIdx.x * blockDim.x + threadIdx.x;
  const int d = idx % DI;
  const int m = idx / DI;
  if (m >= MROWS) return;
  const int l = m & 4095;
  float acc = cb[d];
  const float* base = xz + (size_t)m * (2 * DI) + d;
#pragma unroll
  for (int j = 0; j < 4; ++j) {
    const int t = l - 3 + j;
    if (t >= 0) acc += cw[d * 4 + j] * base[(t - l) * (2 * DI)];
  }
  xc[(size_t)m * DI + d] = acc / (1.f + __expf(-acc));
}

// ---------------------------------------------------------------------------
// K6: selective scan with LDS double-buffered async staging.
// Block = 256 threads = (16 channels) x (16 states); lane n = state index.
// LDS chunk layout (floats): [0,2048)   B|C : t*32 + {n | 16+n}
//                            [2048,3072) dt : t*16 + dl
//                            [3072,4096) x  : t*16 + dl
//                            [4096,5120) z  : t*16 + dl
// ---------------------------------------------------------------------------
__global__ void ssm_scan_kernel(const float* __restrict__ xc,
                                const float* __restrict__ dt,
                                const float* __restrict__ dbc,
                                const float* __restrict__ xz,
                                const float* __restrict__ A_log,
                                const float* __restrict__ Dp,
                                float* __restrict__ yg) {
  __shared__ float smem[2][CHUNK_F];     // 40 KB of the 320 KB WGP LDS

  const int b  = blockIdx.x / (DI / 16);          // 24 blocks per batch
  const int d0 = (blockIdx.x % (DI / 16)) * 16;
  const int tt = threadIdx.x;
  const int dl = tt >> 4;
  const int n  = tt & 15;
  const int d  = d0 + dl;
  const int bL = b * LSEQ;

  const float An = -__expf(A_log[d * NSTATE + n]);
  const float Dd = Dp[d];
  float h = 0.f;

  float* yb = yg + (size_t)bL * DI + d;

  // ---- issue chunk 0 ----
  {
    const int t0 = 0;
    for (int i = tt; i < CHUNK_F; i += 256) {
      const float* g;
      float* l = &smem[0][i];
      if (i < 2048) {
        const int t = i >> 5, c = i & 31;
        g = dbc + (size_t)(bL + t0 + t) * LDDBC + DTRANK + c;
      } else if (i < 3072) {
        const int j = i - 2048, t = j >> 4, c = j & 15;
        g = dt + (size_t)(bL + t0 + t) * DI + d0 + c;
      } else if (i < 4096) {
        const int j = i - 3072, t = j >> 4, c = j & 15;
        g = xc + (size_t)(bL + t0 + t) * DI + d0 + c;
      } else {
        const int j = i - 4096, t = j >> 4, c = j & 15;
        g = xz + (size_t)(bL + t0 + t) * (2 * DI) + DI + d0 + c;
      }
      async_copy_f32(g, l);
    }
  }
  wait_async_all();
  __syncthreads();

  const int NCHUNK = LSEQ / T_CH;        // 64
  for (int ch = 0; ch < NCHUNK; ++ch) {
    const int buf = ch & 1;

    // ---- issue chunk ch+1 into the other buffer (async, no wait yet) ----
    if (ch + 1 < NCHUNK) {
      const int t0 = (ch + 1) * T_CH;
      for (int i = tt; i < CHUNK_F; i += 256) {
        const float* g;
        float* l = &smem[buf ^ 1][i];
        if (i < 2048) {
          const int t = i >> 5, c = i & 31;
          g = dbc + (size_t)(bL + t0 + t) * LDDBC + DTRANK + c;
        } else if (i < 3072) {
          const int j = i - 2048, t = j >> 4, c = j & 15;
          g = dt + (size_t)(bL + t0 + t) * DI + d0 + c;
        } else if (i < 4096) {
          const int j = i - 3072, t = j >> 4, c = j & 15;
          g = xc + (size_t)(bL + t0 + t) * DI + d0 + c;
        } else {
          const int j = i - 4096, t = j >> 4, c = j & 15;
          g = xz + (size_t)(bL + t0 + t) * (2 * DI) + DI + d0 + c;
        }
        async_copy_f32(g, l);
      }
    }

    // ---- consume chunk ch from LDS ----
    const float* s = smem[buf];
    const int tbase = ch * T_CH;
    for (int t = 0; t < T_CH; ++t) {
      const float Bn  = s[t * 32 + n];
      const float Cn  = s[t * 32 + 16 + n];
      const float dtv = s[2048 + t * 16 + dl];
      const float xv  = s[3072 + t * 16 + dl];

      const float dA = __expf(dtv * An);
      h = h * dA + (dtv * xv) * Bn;

      float p = h * Cn;
      p += __shfl_xor(p, 1, 32);
      p += __shfl_xor(p, 2, 32);
      p += __shfl_xor(p, 4, 32);
      p += __shfl_xor(p, 8, 32);

      if (n == 0) {
        const float zv = s[4096 + t * 16 + dl];
        const float y  = (p + Dd * xv) * (zv / (1.f + __expf(-zv)));
        yb[(size_t)(tbase + t) * DI] = y;
      }
    }

    // chunk ch+1 must be resident (and everyone done with buf) before reuse
    wait_async_all();
    __syncthreads();
  }
}

// ---------------------------------------------------------------------------
// K7: out_proj WMMA GEMM + residual, writing [B,C,H,W] directly.
// Wave tile 16x64, N=192 -> 3 n-tiles.
// ---------------------------------------------------------------------------
__global__ void gemm_out_residual_kernel(const float* __restrict__ A,
                                         const float* __restrict__ W,
                                         const float* __restrict__ xin,
                                         float* __restrict__ out) {
  const int wave = blockIdx.x * (blockDim.x >> 5) + (threadIdx.x >> 5);
  const int lane = threadIdx.x & 31;
  const int ntiles = CDIM / 64;   // 3
  const int m0 = (wave / ntiles) * 16;
  const int n0 = (wave % ntiles) * 64;
  if (m0 >= MROWS) return;

  const int hf  = lane >> 4;
  const int l16 = lane & 15;

  v8f acc[4];
#pragma unroll
  for (int j = 0; j < 4; ++j) acc[j] = zero8();

  for (int k0 = 0; k0 < DI; k0 += 4) {
    const int ka = k0 + 2 * hf;
    const v2f afrag = *(const v2f*)(A + (size_t)(m0 + l16) * DI + ka);
#pragma unroll
    for (int j = 0; j < 4; ++j) {
      const int col = n0 + 16 * j + l16;
      const v2f bfrag = *(const v2f*)(W + (size_t)col * DI + ka);
      acc[j] = wmma_f32(afrag, bfrag, acc[j]);
    }
  }

#pragma unroll
  for (int j = 0; j < 4; ++j) {
    const int col = n0 + 16 * j + l16;     // output channel c
#pragma unroll
    for (int r = 0; r < 8; ++r) {
      const int row = m0 + r + 8 * hf;     // (b,l)
      const int b = row >> 12;
      const int l = row & 4095;
      const size_t oi = ((size_t)b * CDIM + col) * LSEQ + l;
      out[oi] = acc[j][r] + xin[oi];       // residual in NCHW layout
    }
  }
}

// ---------------------------------------------------------------------------
extern "C" void kernel_launch(void* const* d_in, const int* in_sizes, int n_in,
                              void* d_out, int out_size, void* d_ws, size_t ws_size,
                              hipStream_t stream) {
  const float* x         = (const float*)d_in[0];
  const float* ln_w      = (const float*)d_in[1];
  const float* ln_b      = (const float*)d_in[2];
  const float* in_proj_w = (const float*)d_in[3];   // [768,192]
  const float* conv_w    = (const float*)d_in[4];   // [384,4]
  const float* conv_b    = (const float*)d_in[5];
  const float* x_proj_w  = (const float*)d_in[6];   // [44,384]
  const float* dt_proj_w = (const float*)d_in[7];   // [384,12]
  const float* dt_proj_b = (const float*)d_in[8];
  const float* A_log     = (const float*)d_in[9];   // [384,16]
  const float* Dp        = (const float*)d_in[10];
  const float* out_proj_w= (const float*)d_in[11];  // [192,384]
  float* out = (float*)d_out;

  float* ws = (float*)d_ws;
  float* XN  = ws;                                  // 16384*192
  float* XZ  = XN  + (size_t)MROWS * CDIM;          // 16384*768
  float* XC  = XZ  + (size_t)MROWS * 2 * DI;        // 16384*384
  float* DT  = XC  + (size_t)MROWS * DI;            // 16384*384
  float* DBC = DT  + (size_t)MROWS * DI;            // 16384*48
  float* YG  = DBC + (size_t)MROWS * LDDBC;         // 16384*384

  // K1: transpose + layernorm (one wave per row, 8 rows/block)
  ln_transpose_kernel<<<MROWS / 8, 256, 0, stream>>>(x, ln_w, ln_b, XN);

  // K2: in_proj  M=16384 N=768 K=192  (1024 m-tiles * 12 n-tiles / 8 waves)
  gemm_wmma_kernel<4, 0><<<1536, 256, 0, stream>>>(
      XN, in_proj_w, XZ, nullptr, MROWS, 2 * DI, CDIM, CDIM, 2 * DI);

  // K3: causal depthwise conv + SiLU
  conv_silu_kernel<<<(MROWS * DI) / 256, 256, 0, stream>>>(XZ, conv_w, conv_b, XC);

  // K4: x_proj  M=16384 N=44 K=384 (padded ldc=48; 1 n-tile of width 48)
  gemm_wmma_kernel<3, 0><<<128, 256, 0, stream>>>(
      XC, x_proj_w, DBC, nullptr, MROWS, DTRANK + 2 * NSTATE, DI, DI, LDDBC);

  // K5: dt_proj + softplus  M=16384 N=384 K=12 (A = DBC cols 0..11, lda=48)
  gemm_wmma_kernel<4, 1><<<768, 256, 0, stream>>>(
      DBC, dt_proj_w, DT, dt_proj_b, MROWS, DI, DTRANK, LDDBC, DI);

  // K6: selective scan, async-LDS double buffered, fused skip + gate
  ssm_scan_kernel<<<B_SZ * (DI / 16), 256, 0, stream>>>(
      XC, DT, DBC, XZ, A_log, Dp, YG);

  // K7: out_proj + residual -> [B,C,H,W]
  gemm_out_residual_kernel<<<384, 256, 0, stream>>>(YG, out_proj_w, x, out);
}